// EdgeDecoder_34376918237201
// MI455X (gfx1250) — compile-verified
//
#include <hip/hip_runtime.h>

typedef __attribute__((ext_vector_type(2))) float v2f;
typedef __attribute__((ext_vector_type(8))) float v8f;

#define D_IN  36
#define D_HID 32
#define D_OUT 32
#define WPITCH 40   // LDS row pitch (floats) for weight tiles: 2*40 % 64 == 16 -> conflict-free
#define XPAD   44   // LDS row pitch (floats) for x/h tiles: 16B aligned, 8*44 % 64 == 32
#define WAVES_PER_BLOCK 8

__global__ __launch_bounds__(256) void edge_mlp_wmma(
    const float* __restrict__ src, const float* __restrict__ dst,
    const float* __restrict__ u,   const int*   __restrict__ batch,
    const float* __restrict__ W1,  const float* __restrict__ b1,
    const float* __restrict__ W2,  const float* __restrict__ b2,
    float* __restrict__ out, int E)
{
    __shared__ float sW1[D_IN  * WPITCH];
    __shared__ float sW2[D_HID * WPITCH];
    __shared__ float sb1[D_HID];
    __shared__ float sb2[D_OUT];
    __shared__ float sX[WAVES_PER_BLOCK][16 * XPAD];

    const int tid = threadIdx.x;

    // Stage weights once per block (padded pitch for bank-conflict-free reads).
    for (int i = tid; i < D_IN * D_HID; i += blockDim.x) {
        int k = i >> 5, n = i & 31;
        sW1[k * WPITCH + n] = W1[i];
    }
    for (int i = tid; i < D_HID * D_OUT; i += blockDim.x) {
        int k = i >> 5, n = i & 31;
        sW2[k * WPITCH + n] = W2[i];
    }
    if (tid < D_HID)                 sb1[tid] = b1[tid];
    else if (tid < D_HID + D_OUT)    sb2[tid - D_HID] = b2[tid - D_HID];
    __syncthreads();

    const int lane = tid & 31;
    const int wib  = tid >> 5;
    const int r    = lane & 15;   // tile row for A / column for B,C,D
    const int half = lane >> 4;   // lane half selects K pair / M upper half
    float* xs = sX[wib];

    const float bias10 = sb1[r];
    const float bias11 = sb1[16 + r];
    const float bias20 = sb2[r];
    const float bias21 = sb2[16 + r];

    const int  numTiles = (E + 15) >> 4;
    const long gwave0   = (long)blockIdx.x * WAVES_PER_BLOCK + wib;
    const long nwaves   = (long)gridDim.x * WAVES_PER_BLOCK;

    for (long t = gwave0; t < numTiles; t += nwaves) {
        const long e0 = t << 4;
        long e  = e0 + r;
        long el = (e < (long)E - 1) ? e : ((long)E - 1);  // clamp (tail safety)

        // ---- stage x tile: [src(2) | dest(2) | u[batch[e]](32)] per row ----
        if (half == 0) {
            const float2 sv = ((const float2*)src)[el];
            xs[r * XPAD + 0] = sv.x;
            xs[r * XPAD + 1] = sv.y;
        } else {
            const float2 dv = ((const float2*)dst)[el];
            xs[r * XPAD + 2] = dv.x;
            xs[r * XPAD + 3] = dv.y;
        }
        const int g = batch[el];
        const float4* up = (const float4*)(u + (size_t)g * 32 + half * 16);
        float4* xp = (float4*)(xs + r * XPAD + 4 + half * 16);
        xp[0] = up[0]; xp[1] = up[1]; xp[2] = up[2]; xp[3] = up[3];
        // intra-wave LDS ordering (DScnt) makes produced data visible; no barrier needed

        // ---- GEMM1: h = relu(x @ W1 + b1), K = 36 in 9 steps of 4 ----
        v8f acc0 = {};
        v8f acc1 = {};
#pragma unroll
        for (int k0 = 0; k0 < D_IN; k0 += 4) {
            const int c = k0 + 2 * half;
            v2f a;
            a.x = xs[r * XPAD + c];
            a.y = xs[r * XPAD + c + 1];
            v2f bA, bB;
            bA.x = sW1[c * WPITCH + r];
            bA.y = sW1[(c + 1) * WPITCH + r];
            bB.x = sW1[c * WPITCH + 16 + r];
            bB.y = sW1[(c + 1) * WPITCH + 16 + r];
            acc0 = __builtin_amdgcn_wmma_f32_16x16x4_f32(false, a, false, bA,
                                                         (short)0, acc0, false, false);
            acc1 = __builtin_amdgcn_wmma_f32_16x16x4_f32(false, a, false, bB,
                                                         (short)0, acc1, false, false);
        }

        // ---- bias + ReLU, transpose h back into LDS (C-layout -> row major) ----
#pragma unroll
        for (int i = 0; i < 8; i++) {
            const int m = i + 8 * half;  // C/D layout: lanes 16-31 hold M = 8 + vgpr
            xs[m * XPAD + r]      = fmaxf(acc0[i] + bias10, 0.0f);
            xs[m * XPAD + 16 + r] = fmaxf(acc1[i] + bias11, 0.0f);
        }

        // ---- GEMM2: o = h @ W2 + b2, K = 32 in 8 steps of 4 ----
        v8f o0 = {};
        v8f o1 = {};
#pragma unroll
        for (int k0 = 0; k0 < D_HID; k0 += 4) {
            const int c = k0 + 2 * half;
            v2f a;
            a.x = xs[r * XPAD + c];
            a.y = xs[r * XPAD + c + 1];
            v2f bA, bB;
            bA.x = sW2[c * WPITCH + r];
            bA.y = sW2[(c + 1) * WPITCH + r];
            bB.x = sW2[c * WPITCH + 16 + r];
            bB.y = sW2[(c + 1) * WPITCH + 16 + r];
            o0 = __builtin_amdgcn_wmma_f32_16x16x4_f32(false, a, false, bA,
                                                       (short)0, o0, false, false);
            o1 = __builtin_amdgcn_wmma_f32_16x16x4_f32(false, a, false, bB,
                                                       (short)0, o1, false, false);
        }

        // ---- store: per vgpr i, lanes 0-15 -> row e0+i, lanes 16-31 -> row e0+i+8 ----
#pragma unroll
        for (int i = 0; i < 8; i++) {
            const long m = e0 + i + 8 * half;
            if (m < E) {
                out[m * 32 + r]      = o0[i] + bias20;
                out[m * 32 + 16 + r] = o1[i] + bias21;
            }
        }
    }
}

extern "C" void kernel_launch(void* const* d_in, const int* in_sizes, int n_in,
                              void* d_out, int out_size, void* d_ws, size_t ws_size,
                              hipStream_t stream) {
    const float* src   = (const float*)d_in[0];
    const float* dst   = (const float*)d_in[1];
    // d_in[2] = edge_attr: present in inputs but unused by the reference
    const float* u     = (const float*)d_in[3];
    const int*   batch = (const int*)d_in[4];
    const float* W1    = (const float*)d_in[5];
    const float* b1    = (const float*)d_in[6];
    const float* W2    = (const float*)d_in[7];
    const float* b2    = (const float*)d_in[8];
    float* out = (float*)d_out;

    const int E = in_sizes[0] / 2;              // src is [E, 2]
    const int numTiles = (E + 15) / 16;
    int blocksNeeded = (numTiles + WAVES_PER_BLOCK - 1) / WAVES_PER_BLOCK;
    int blocks = blocksNeeded < 2048 ? blocksNeeded : 2048;  // grid-stride beyond this

    edge_mlp_wmma<<<blocks, 256, 0, stream>>>(src, dst, u, batch, W1, b1, W2, b2, out, E);
}